// Decoder_60378650247791
// MI455X (gfx1250) — compile-verified
//
#include <hip/hip_runtime.h>
#include <hip/hip_bf16.h>
#include <math.h>

// Shapes from the reference
#define VOCAB_N 32000
#define EMB_N   256
#define UNITS_N 1024
#define B_N     64
#define T_N     128

typedef float v2f __attribute__((ext_vector_type(2)));
typedef float v8f __attribute__((ext_vector_type(8)));

#define LDA 68  // padded LDS stride for A tile: avoids ds bank conflicts on
                // b64 A-fragment reads (banks = 4r+ka+{0..3}, all distinct)
#define LDB 64  // B tile stride; b-frag reads are lane-contiguous already

// ---------------------------------------------------------------------------
// Tiled fp32 WMMA GEMM:  C(M x N) = A(M x K, row-major) * B(K x N, row-major)
// Block = 128 threads (4 waves). Block tile = 64 (M) x 64 (N).
// Wave w owns columns [w*16, w*16+16) of the tile and ALL 64 rows via four
// 16x16 accumulator sets. K is chunked by 64; both A and B chunks are staged
// in LDS with float4 loads, so the inner loop is ds_load + v_wmma only.
//
// V_WMMA_F32_16X16X4_F32 fragment layouts (ISA 7.12.2):
//   A (16x4):  lane L (r=L&15, h=L>>4): v0=A[r][k+2h], v1=A[r][k+2h+1]
//   B (4x16):  lane L:                  v0=B[k+2h][n],  v1=B[k+2h+1][n]
//   C (16x16): vgpr i, lane L: element (m = i + 8h, n = n0+r)
//
// MODE 0: out[m][n] = acc + bias[n]
// MODE 1: attention-score epilogue:
//         score[m] += sum_n V[n] * tanh(acc + bias[n] + rowAdd[m>>7][n])
// ---------------------------------------------------------------------------
template <int MODE>
__global__ __launch_bounds__(128) void gemm_wmma_f32(
    const float* __restrict__ A, const float* __restrict__ B,
    const float* __restrict__ bias, float* __restrict__ out, int K, int N,
    const float* __restrict__ rowAdd,  // MODE 1: hW2 (B_N x N)
    const float* __restrict__ Vvec,    // MODE 1: V (N)
    float* __restrict__ score)         // MODE 1: (M) accumulator
{
  __shared__ __align__(16) float As[64 * LDA];  // 17 KB
  __shared__ __align__(16) float Bs[64 * LDB];  // 16 KB

  const int tid  = threadIdx.x;
  const int lane = tid & 31;
  const int wave = tid >> 5;
  const int h = lane >> 4;  // half-wave: 0 or 1
  const int r = lane & 15;
  const int m0 = blockIdx.x * 64;
  const int nblk = blockIdx.y * 64;
  const int nl = wave * 16 + r;  // column within block tile
  const int n  = nblk + nl;      // global column

  // Tile-load coordinates: thread t loads rows (t>>4)+8i, cols (t&15)*4..+3
  const int lrow = tid >> 4;         // 0..7
  const int lcol = (tid & 15) << 2;  // 0,4,...,60
  const float* ap = A + (size_t)(m0 + lrow) * K + lcol;
  const float* bp = B + (size_t)lrow * N + nblk + lcol;

  v8f acc[4] = {};

  for (int kb = 0; kb < K; kb += 64) {
    const float* bpk = bp + (size_t)kb * N;
    const float* apk = ap + kb;
    const size_t strideA = (size_t)8 * K;
    const size_t strideB = (size_t)8 * N;
#pragma unroll
    for (int i = 0; i < 8; ++i) {
      *(float4*)&As[(lrow + 8 * i) * LDA + lcol] =
          *(const float4*)(apk + strideA * i);
      *(float4*)&Bs[(lrow + 8 * i) * LDB + lcol] =
          *(const float4*)(bpk + strideB * i);
    }
    __syncthreads();

#pragma unroll
    for (int kk = 0; kk < 64; kk += 4) {
      const int ka = kk + 2 * h;
      v2f bfrag;
      bfrag.x = Bs[ka * LDB + nl];
      bfrag.y = Bs[(ka + 1) * LDB + nl];
#pragma unroll
      for (int mi = 0; mi < 4; ++mi) {
        v2f afrag = *(const v2f*)&As[(mi * 16 + r) * LDA + ka];
        acc[mi] = __builtin_amdgcn_wmma_f32_16x16x4_f32(
            false, afrag, false, bfrag, (short)0, acc[mi], false, false);
      }
    }
    __syncthreads();
  }

  if constexpr (MODE == 0) {
    const float bn = bias[n];
#pragma unroll
    for (int mi = 0; mi < 4; ++mi) {
#pragma unroll
      for (int i = 0; i < 8; ++i) {
        const int m = m0 + mi * 16 + i + 8 * h;
        out[(size_t)m * N + n] = acc[mi][i] + bn;
      }
    }
  } else {
    const float bn = bias[n];
    const float vn = Vvec[n];
#pragma unroll
    for (int mi = 0; mi < 4; ++mi) {
#pragma unroll
      for (int i = 0; i < 8; ++i) {
        const int m = m0 + mi * 16 + i + 8 * h;
        const int b = m >> 7;  // T_N == 128
        float val = tanhf(acc[mi][i] + bn + rowAdd[(size_t)b * N + n]) * vn;
        // reduce over the 16 lanes of this half-wave (xor offsets < 16 stay
        // inside the half)
        for (int off = 8; off; off >>= 1) val += __shfl_xor(val, off, 32);
        if (r == 0) atomicAdd(&score[m], val);
      }
    }
  }
}

// score[i] = bV  (bias of the attention scorer, added before softmax)
__global__ void init_score(float* __restrict__ score,
                           const float* __restrict__ bV) {
  int i = blockIdx.x * blockDim.x + threadIdx.x;
  if (i < B_N * T_N) score[i] = bV[0];
}

// Per-batch softmax over T, attention weights out, context + embedding gather
// packed into gin = [context (1024) | emb[x[b]] (256)].
__global__ __launch_bounds__(T_N) void attn_softmax_context(
    const float* __restrict__ score, const float* __restrict__ enc,
    const float* __restrict__ emb, const int* __restrict__ x,
    float* __restrict__ attn_out, float* __restrict__ gin) {
  const int b = blockIdx.x;
  const int t = threadIdx.x;  // 128 threads = T_N
  __shared__ float w[T_N];
  __shared__ float part[4];

  float s = score[b * T_N + t];

  // block max
  float m = s;
  for (int off = 16; off; off >>= 1) m = fmaxf(m, __shfl_xor(m, off, 32));
  if ((t & 31) == 0) part[t >> 5] = m;
  __syncthreads();
  m = fmaxf(fmaxf(part[0], part[1]), fmaxf(part[2], part[3]));
  __syncthreads();

  float e = expf(s - m);
  float sum = e;
  for (int off = 16; off; off >>= 1) sum += __shfl_xor(sum, off, 32);
  if ((t & 31) == 0) part[t >> 5] = sum;
  __syncthreads();
  sum = part[0] + part[1] + part[2] + part[3];

  const float wv = e / sum;
  w[t] = wv;
  attn_out[b * T_N + t] = wv;
  __syncthreads();

  // context[b][k] = sum_t w[t] * enc[b][t][k]
  for (int k = t; k < UNITS_N; k += T_N) {
    float c = 0.f;
    const float* ep = enc + ((size_t)b * T_N) * UNITS_N + k;
#pragma unroll 8
    for (int tt = 0; tt < T_N; ++tt) c += w[tt] * ep[(size_t)tt * UNITS_N];
    gin[(size_t)b * (UNITS_N + EMB_N) + k] = c;
  }
  // embedding gather
  const int tok = x[b];
  for (int e2 = t; e2 < EMB_N; e2 += T_N)
    gin[(size_t)b * (UNITS_N + EMB_N) + UNITS_N + e2] =
        emb[(size_t)tok * EMB_N + e2];
}

// GRU with h0 == 0:  state = (1 - sigmoid(gx_z)) * tanh(gx_n)
__global__ void gru_state(const float* __restrict__ gates,
                          float* __restrict__ state) {
  int i = blockIdx.x * blockDim.x + threadIdx.x;  // B_N * UNITS_N
  if (i >= B_N * UNITS_N) return;
  const int b = i >> 10, j = i & (UNITS_N - 1);
  const float gz = gates[(size_t)b * 3 * UNITS_N + j];
  const float gn = gates[(size_t)b * 3 * UNITS_N + 2 * UNITS_N + j];
  const float z = 1.f / (1.f + expf(-gz));
  state[i] = (1.f - z) * tanhf(gn);
}

extern "C" void kernel_launch(void* const* d_in, const int* in_sizes, int n_in,
                              void* d_out, int out_size, void* d_ws,
                              size_t ws_size, hipStream_t stream) {
  (void)in_sizes; (void)n_in; (void)out_size; (void)ws_size;

  const int*   x      = (const int*)d_in[0];
  const float* hidden = (const float*)d_in[1];
  const float* enc    = (const float*)d_in[2];
  const float* emb    = (const float*)d_in[3];
  const float* W1     = (const float*)d_in[4];
  const float* b1     = (const float*)d_in[5];
  const float* W2     = (const float*)d_in[6];
  const float* b2     = (const float*)d_in[7];
  const float* V      = (const float*)d_in[8];
  const float* bV     = (const float*)d_in[9];
  const float* gru_k  = (const float*)d_in[10];
  // d_in[11] = gru_rk: dead (h0 == 0)
  const float* gru_b  = (const float*)d_in[12];
  const float* fc_W   = (const float*)d_in[13];
  const float* fc_b   = (const float*)d_in[14];

  // Output layout: logits (64x32000) | state (64x1024) | attn (64x128)
  float* logits = (float*)d_out;
  float* state  = logits + (size_t)B_N * VOCAB_N;
  float* attn   = state + (size_t)B_N * UNITS_N;

  // Workspace carve-up (~1.4 MB)
  float* score = (float*)d_ws;                       // 64*128
  float* hw2   = score + B_N * T_N;                  // 64*1024
  float* gin   = hw2 + B_N * UNITS_N;                // 64*1280
  float* gates = gin + B_N * (UNITS_N + EMB_N);      // 64*3072

  // 0) score <- bV
  init_score<<<dim3((B_N * T_N + 255) / 256), dim3(256), 0, stream>>>(score, bV);

  // 1) hW2 = hidden @ W2 + b2             (64x1024 @ 1024x1024)
  gemm_wmma_f32<0><<<dim3(1, UNITS_N / 64), dim3(128), 0, stream>>>(
      hidden, W2, b2, hw2, UNITS_N, UNITS_N, nullptr, nullptr, nullptr);

  // 2) score += sum_n V[n]*tanh(enc@W1 + b1 + hW2)   (8192x1024 @ 1024x1024)
  gemm_wmma_f32<1><<<dim3(B_N * T_N / 64, UNITS_N / 64), dim3(128), 0, stream>>>(
      enc, W1, b1, nullptr, UNITS_N, UNITS_N, hw2, V, score);

  // 3) softmax over T, attention weights, context, embedding gather -> gin
  attn_softmax_context<<<dim3(B_N), dim3(T_N), 0, stream>>>(score, enc, emb, x,
                                                            attn, gin);

  // 4) gates = gin @ gru_k + gru_b        (64x1280 @ 1280x3072)
  gemm_wmma_f32<0><<<dim3(1, 3 * UNITS_N / 64), dim3(128), 0, stream>>>(
      gin, gru_k, gru_b, gates, UNITS_N + EMB_N, 3 * UNITS_N, nullptr, nullptr,
      nullptr);

  // 5) state = (1 - sigmoid(z)) * tanh(n)
  gru_state<<<dim3((B_N * UNITS_N + 255) / 256), dim3(256), 0, stream>>>(gates,
                                                                         state);

  // 6) logits = state @ fc_W + fc_b       (64x1024 @ 1024x32000)
  gemm_wmma_f32<0><<<dim3(1, VOCAB_N / 64), dim3(128), 0, stream>>>(
      state, fc_W, fc_b, logits, UNITS_N, VOCAB_N, nullptr, nullptr, nullptr);
}